// IFA_MatchingNet_65730179498741
// MI455X (gfx1250) — compile-verified
//
#include <hip/hip_runtime.h>

#define BSZ  16
#define CSZ  512
#define HWSZ 3600
#define EPSF 1e-8f

typedef __attribute__((ext_vector_type(2))) float v2f;
typedef __attribute__((ext_vector_type(8))) float v8f;

__device__ __forceinline__ unsigned fenc(float f) {
    unsigned u = __float_as_uint(f);
    return (u & 0x80000000u) ? ~u : (u | 0x80000000u);
}
__device__ __forceinline__ float fdec(unsigned e) {
    unsigned u = (e & 0x80000000u) ? (e & 0x7FFFFFFFu) : ~e;
    return __uint_as_float(u);
}

// ---------------------------------------------------------------------------
// Per-batch L2 norms of two C-vectors: n = max(sqrt(sum v^2), EPS)
// grid = B, block = 256
__global__ void knorm2(const float* __restrict__ A, const float* __restrict__ Bv,
                       float* __restrict__ nA, float* __restrict__ nB) {
    const int b = blockIdx.x, t = threadIdx.x;
    const float* a  = A  + b * CSZ;
    const float* bb = Bv + b * CSZ;
    float sa = 0.f, sb = 0.f;
    for (int c = t; c < CSZ; c += 256) {
        float x = a[c];  sa += x * x;
        float y = bb[c]; sb += y * y;
    }
    __shared__ float ra[256], rb[256];
    ra[t] = sa; rb[t] = sb; __syncthreads();
    for (int s = 128; s > 0; s >>= 1) {
        if (t < s) { ra[t] += ra[t + s]; rb[t] += rb[t + s]; }
        __syncthreads();
    }
    if (t == 0) {
        nA[b] = fmaxf(sqrtf(ra[0]), EPSF);
        nB[b] = fmaxf(sqrtf(rb[0]), EPSF);
    }
}

// ---------------------------------------------------------------------------
// WMMA cosine stage. One wave computes 16 pixels: D = A(16x4) x B(4x16)
// accumulated over K = C = 512 in steps of 4 using V_WMMA_F32_16X16X4_F32.
// A rows: M=0 -> vecA, M=1 -> vecB, rest zero. Vectors staged in LDS as
// interleaved float2 so A-operand fetch is an unconditional ds_load_b64
// broadcast + v_cndmask (no EXEC-masked loads on the critical path).
// MODE 0: also accumulate per-pixel ||q||^2; emit pred_fg, qnorm.
// MODE 1: read qnorm; emit act = dotA/(qn*nA), deact = dotB/(qn*nB).
// grid = B*29, block = 256 (8 waves, 8 pixel-tiles of 16; 225 tiles/batch)
template <int MODE>
__global__ void kwmma(const float* __restrict__ q,
                      const float* __restrict__ vA, const float* __restrict__ vB,
                      const float* __restrict__ nA, const float* __restrict__ nB,
                      float* __restrict__ qnorm,
                      float* __restrict__ out0, float* __restrict__ out1) {
    __shared__ v2f sAB[CSZ];
    const int b   = blockIdx.x / 29;
    const int blk = blockIdx.x % 29;
    for (int c = threadIdx.x; c < CSZ; c += 256) {
        v2f t; t.x = vA[b * CSZ + c]; t.y = vB[b * CSZ + c];
        sAB[c] = t;
    }
    __syncthreads();

    const int wave = threadIdx.x >> 5;
    const int lane = threadIdx.x & 31;
    const int tile = blk * 8 + wave;
    if (tile >= HWSZ / 16) return;   // wave-uniform exit; EXEC stays all-1 for WMMA

    const int p0   = tile * 16;
    const int half = lane >> 4;      // 0: lanes 0-15, 1: lanes 16-31
    const int n    = lane & 15;      // pixel-within-tile (B column), also A-row M
    const bool selA = (n == 0);
    const bool selB = (n == 1);

    v8f acc = {0.f, 0.f, 0.f, 0.f, 0.f, 0.f, 0.f, 0.f};
    float ss = 0.f;
    const float* qbase = q + (size_t)b * CSZ * HWSZ + (size_t)(2 * half) * HWSZ + p0 + n;

    #pragma unroll 4
    for (int kk = 0; kk < CSZ; kk += 4) {
        const int kA = kk + 2 * half;                 // K = 2*half + j, j in {0,1}
        const v2f pr0 = sAB[kA];                      // {vA[kA],   vB[kA]}   broadcast b64
        const v2f pr1 = sAB[kA + 1];                  // {vA[kA+1], vB[kA+1]}
        v2f a;
        a.x = selA ? pr0.x : (selB ? pr0.y : 0.f);
        a.y = selA ? pr1.x : (selB ? pr1.y : 0.f);
        const float* qp = qbase + (size_t)kk * HWSZ;
        v2f bb;
        bb.x = qp[0];
        bb.y = qp[HWSZ];
        acc = __builtin_amdgcn_wmma_f32_16x16x4_f32(false, a, false, bb,
                                                    (short)0, acc, false, false);
        if (MODE == 0) ss += bb.x * bb.x + bb.y * bb.y;
    }

    const float dA = acc[0];  // D[M=0][n], valid lanes 0-15
    const float dB = acc[1];  // D[M=1][n], valid lanes 0-15

    if (MODE == 0) {
        float s2 = ss + __shfl_xor(ss, 16, 32);       // combine lane halves -> sum_k q^2
        float qn = fmaxf(sqrtf(s2), EPSF);
        float simA = dA / (qn * nA[b]);
        float simB = dB / (qn * nB[b]);
        float pred = 1.f / (1.f + __expf(-10.f * (simA - simB)));  // softmax fg channel
        if (lane < 16) {
            qnorm[b * HWSZ + p0 + n] = qn;
            out0 [b * HWSZ + p0 + n] = pred;
        }
    } else {
        if (lane < 16) {
            float qn = qnorm[b * HWSZ + p0 + n];
            out0[b * HWSZ + p0 + n] = dA / (qn * nA[b]);
            out1[b * HWSZ + p0 + n] = dB / (qn * nB[b]);
        }
    }
}

// ---------------------------------------------------------------------------
// Threshold masks + top-12 fallback + weight sums. grid = B, block = 256.
__global__ void kweights(const float* __restrict__ pred_fg, const float* __restrict__ tau,
                         float* __restrict__ wfg, float* __restrict__ wbg,
                         float* __restrict__ wsfg, float* __restrict__ wsbg) {
    const int b = blockIdx.x, t = threadIdx.x;
    const float* p = pred_fg + b * HWSZ;
    const float thres = 1.f / (1.f + __expf(-tau[0]));   // sigmoid(tau)

    __shared__ int   ri[256];
    __shared__ float rv[256];

    int cfg = 0, cbg = 0;
    for (int i = t; i < HWSZ; i += 256) {
        float v = p[i];
        cfg += (v > thres) ? 1 : 0;
        cbg += ((1.f - v) > (1.f - thres)) ? 1 : 0;
    }
    ri[t] = cfg | (cbg << 16);
    __syncthreads();
    for (int s = 128; s > 0; s >>= 1) {
        if (t < s) ri[t] += ri[t + s];
        __syncthreads();
    }
    const int cfgT = ri[0] & 0xFFFF;
    const int cbgT = ri[0] >> 16;
    __syncthreads();

    // ---- FG weights ----
    if (cfgT > 0) {
        for (int i = t; i < HWSZ; i += 256) wfg[b * HWSZ + i] = (p[i] > thres) ? 1.f : 0.f;
    } else {
        for (int i = t; i < HWSZ; i += 256) wfg[b * HWSZ + i] = 0.f;
        __syncthreads();
        for (int it = 0; it < 12; ++it) {            // iterative argmax, first-index ties
            float bv = -1e30f; int bi = HWSZ;
            for (int i = t; i < HWSZ; i += 256) {
                if (wfg[b * HWSZ + i] == 0.f) {
                    float v = p[i];
                    if (v > bv || (v == bv && i < bi)) { bv = v; bi = i; }
                }
            }
            rv[t] = bv; ri[t] = bi; __syncthreads();
            for (int s = 128; s > 0; s >>= 1) {
                if (t < s) {
                    float v2 = rv[t + s]; int i2 = ri[t + s];
                    if (v2 > rv[t] || (v2 == rv[t] && i2 < ri[t])) { rv[t] = v2; ri[t] = i2; }
                }
                __syncthreads();
            }
            if (t == 0) wfg[b * HWSZ + ri[0]] = 1.f;
            __syncthreads();
        }
    }
    __syncthreads();

    // ---- BG weights ----
    if (cbgT > 0) {
        for (int i = t; i < HWSZ; i += 256)
            wbg[b * HWSZ + i] = ((1.f - p[i]) > (1.f - thres)) ? 1.f : 0.f;
    } else {
        for (int i = t; i < HWSZ; i += 256) wbg[b * HWSZ + i] = 0.f;
        __syncthreads();
        for (int it = 0; it < 12; ++it) {
            float bv = -1e30f; int bi = HWSZ;
            for (int i = t; i < HWSZ; i += 256) {
                if (wbg[b * HWSZ + i] == 0.f) {
                    float v = 1.f - p[i];
                    if (v > bv || (v == bv && i < bi)) { bv = v; bi = i; }
                }
            }
            rv[t] = bv; ri[t] = bi; __syncthreads();
            for (int s = 128; s > 0; s >>= 1) {
                if (t < s) {
                    float v2 = rv[t + s]; int i2 = ri[t + s];
                    if (v2 > rv[t] || (v2 == rv[t] && i2 < ri[t])) { rv[t] = v2; ri[t] = i2; }
                }
                __syncthreads();
            }
            if (t == 0) wbg[b * HWSZ + ri[0]] = 1.f;
            __syncthreads();
        }
    }

    if (t == 0) {
        wsfg[b] = (cfgT > 0) ? (float)cfgT : 12.f;
        wsbg[b] = (cbgT > 0) ? (float)cbgT : 12.f;
    }
}

// ---------------------------------------------------------------------------
// Prototypes: proto[b][c] = sum_p q[b][c][p]*w[b][p] / wsum[b].
// grid = B * (C/8); block = 256 (8 waves, one c-row per wave); weights in LDS.
__global__ void kproto(const float* __restrict__ q,
                       const float* __restrict__ wfg, const float* __restrict__ wbg,
                       const float* __restrict__ wsfg, const float* __restrict__ wsbg,
                       float* __restrict__ outFP, float* __restrict__ outBP) {
    __shared__ float swf[HWSZ];
    __shared__ float swb[HWSZ];
    const int b  = blockIdx.x >> 6;         // 64 blocks per batch
    const int c0 = (blockIdx.x & 63) * 8;
    for (int i = threadIdx.x; i < HWSZ; i += 256) {
        swf[i] = wfg[b * HWSZ + i];
        swb[i] = wbg[b * HWSZ + i];
    }
    __syncthreads();
    const int wave = threadIdx.x >> 5, lane = threadIdx.x & 31;
    const int c = c0 + wave;
    const float* row = q + (size_t)(b * CSZ + c) * HWSZ;
    float sf = 0.f, sb = 0.f;
    for (int i = lane; i < HWSZ; i += 32) {
        float v = row[i];
        sf += v * swf[i];
        sb += v * swb[i];
    }
    for (int off = 16; off > 0; off >>= 1) {
        sf += __shfl_down(sf, off, 32);
        sb += __shfl_down(sb, off, 32);
    }
    if (lane == 0) {
        outFP[b * CSZ + c] = sf / wsfg[b];
        outBP[b * CSZ + c] = sb / wsbg[b];
    }
}

// ---------------------------------------------------------------------------
__global__ void kmmInit(unsigned* mm) {
    if (threadIdx.x == 0) { mm[0] = 0xFFFFFFFFu; mm[1] = 0u; mm[2] = 0xFFFFFFFFu; mm[3] = 0u; }
}

// Global min/max of act & deact (ordered-uint encoding). grid = 225, block = 256.
__global__ void kminmax(const float* __restrict__ act, const float* __restrict__ deact,
                        unsigned* __restrict__ mm) {
    const int t = threadIdx.x;
    const int i = blockIdx.x * 256 + t;      // exactly covers B*HW = 57600
    unsigned ea = fenc(act[i]), ed = fenc(deact[i]);
    __shared__ unsigned s0[256], s1[256], s2[256], s3[256];
    s0[t] = ea; s1[t] = ea; s2[t] = ed; s3[t] = ed;
    __syncthreads();
    for (int s = 128; s > 0; s >>= 1) {
        if (t < s) {
            s0[t] = min(s0[t], s0[t + s]);
            s1[t] = max(s1[t], s1[t + s]);
            s2[t] = min(s2[t], s2[t + s]);
            s3[t] = max(s3[t], s3[t + s]);
        }
        __syncthreads();
    }
    if (t == 0) {
        atomicMin(&mm[0], s0[0]); atomicMax(&mm[1], s1[0]);
        atomicMin(&mm[2], s2[0]); atomicMax(&mm[3], s3[0]);
    }
}

// factor[b][p] = norm(act) + 1 - norm(deact). grid = 225, block = 256.
__global__ void kfactor(const float* __restrict__ act, const float* __restrict__ deact,
                        const unsigned* __restrict__ mm, float* __restrict__ factor) {
    const int i = blockIdx.x * 256 + threadIdx.x;
    const float amin = fdec(mm[0]), amax = fdec(mm[1]);
    const float dmin = fdec(mm[2]), dmax = fdec(mm[3]);
    float an = (act[i]   - amin) / (amax - amin);
    float dn = (deact[i] - dmin) / (dmax - dmin);
    factor[i] = an + 1.f - dn;
}

// q_out = q * factor[b][p], float4 streamed both sides. grid = 28800, block = 256.
__global__ void kfinal(const float* __restrict__ q, const float* __restrict__ factor,
                       float* __restrict__ out) {
    const size_t g = (size_t)blockIdx.x * 256 + threadIdx.x;  // 7,372,800 groups of 4
    float4 v = ((const float4*)q)[g];
    const size_t e = g * 4;
    const int b = (int)(e / ((size_t)CSZ * HWSZ));
    const int p = (int)(e % HWSZ);                            // p % 4 == 0 -> 16B aligned
    float4 f = *(const float4*)(factor + b * HWSZ + p);
    v.x *= f.x; v.y *= f.y; v.z *= f.z; v.w *= f.w;
    ((float4*)out)[g] = v;
}

// ---------------------------------------------------------------------------
extern "C" void kernel_launch(void* const* d_in, const int* in_sizes, int n_in,
                              void* d_out, int out_size, void* d_ws, size_t ws_size,
                              hipStream_t stream) {
    const float* supp_fp = (const float*)d_in[0];   // B*C
    const float* supp_bp = (const float*)d_in[1];   // B*C
    const float* q       = (const float*)d_in[2];   // B*C*HW
    const float* tau     = (const float*)d_in[3];   // 1

    float* out   = (float*)d_out;
    float* outFP = out + (size_t)BSZ * CSZ * HWSZ;  // query_fp output (B*C)
    float* outBP = outFP + BSZ * CSZ;               // query_bp output (B*C)

    float* ws     = (float*)d_ws;
    float* qnorm  = ws;                 // B*HW
    float* pred   = ws + 57600;         // B*HW
    float* wfg    = ws + 115200;        // B*HW
    float* wbg    = ws + 172800;        // B*HW
    float* act    = ws + 230400;        // B*HW
    float* deact  = ws + 288000;        // B*HW
    float* factor = ws + 345600;        // B*HW
    float* wsfg   = ws + 403200;        // B
    float* wsbg   = ws + 403216;        // B
    float* nfp    = ws + 403232;        // B
    float* nbp    = ws + 403248;        // B
    float* npf    = ws + 403264;        // B
    float* npb    = ws + 403280;        // B
    unsigned* mm  = (unsigned*)(ws + 403296);       // 4

    knorm2 <<<BSZ,      256, 0, stream>>>(supp_fp, supp_bp, nfp, nbp);
    kwmma<0><<<BSZ * 29, 256, 0, stream>>>(q, supp_fp, supp_bp, nfp, nbp, qnorm, pred, nullptr);
    kweights<<<BSZ,     256, 0, stream>>>(pred, tau, wfg, wbg, wsfg, wsbg);
    kproto <<<BSZ * 64, 256, 0, stream>>>(q, wfg, wbg, wsfg, wsbg, outFP, outBP);
    knorm2 <<<BSZ,      256, 0, stream>>>(outFP, outBP, npf, npb);
    kwmma<1><<<BSZ * 29, 256, 0, stream>>>(q, outFP, outBP, npf, npb, qnorm, act, deact);
    kmmInit<<<1,         32, 0, stream>>>(mm);
    kminmax<<<225,      256, 0, stream>>>(act, deact, mm);
    kfactor<<<225,      256, 0, stream>>>(act, deact, mm, factor);
    kfinal <<<28800,    256, 0, stream>>>(q, factor, out);
}